// FM_model_30408368456213
// MI455X (gfx1250) — compile-verified
//
#include <hip/hip_runtime.h>
#include <math.h>

// ---------------- FM model geometry ----------------
#define VL        32
#define ROW       33                 // VL embedding floats + 1 linear float
#define NF_USER   20
#define NF_CTX    10
#define NF_DOC    30
#define NF        60                 // total fields
#define SAMPLE_FLOATS (NF * ROW)     // 1980 floats = 7920 B per sample

#define WAVES_PER_BLOCK 4
#define BLOCK_THREADS   (WAVES_PER_BLOCK * 32)

// ---------------- CDNA5 async global->LDS path ----------------
#ifndef __has_builtin
#define __has_builtin(x) 0
#endif

#if __has_builtin(__builtin_amdgcn_global_load_async_to_lds_b32)
#define HAVE_ASYNC_B32 1
#else
#define HAVE_ASYNC_B32 0
#endif

#if __has_builtin(__builtin_amdgcn_s_wait_asynccnt)
#define HAVE_WAIT_ASYNC 1
#else
#define HAVE_WAIT_ASYNC 0
#endif

typedef __attribute__((address_space(1))) int* gas_iptr; // global, as builtin wants
typedef __attribute__((address_space(3))) int* las_iptr; // LDS,    as builtin wants

// One async 4B/lane copy at compile-time byte offset OFF from both addresses:
// the same immediate offset is added to the LDS and the global address.
template <int OFF>
__device__ __forceinline__ void async_g2l_b32_off(const float* g, float* l) {
#if HAVE_ASYNC_B32
  __builtin_amdgcn_global_load_async_to_lds_b32(
      (gas_iptr)(int*)(size_t)g, (las_iptr)(int*)l, OFF, /*cpol*/0);
#else
  unsigned lds_addr = (unsigned)(unsigned long long)(las_iptr)(int*)l;
  asm volatile("global_load_async_to_lds_b32 %0, %1, off offset:%2"
               :
               : "v"(lds_addr), "v"(g), "i"(OFF)
               : "memory");
#endif
}

__device__ __forceinline__ void wait_async0() {
#if HAVE_WAIT_ASYNC
  __builtin_amdgcn_s_wait_asynccnt(0);
  asm volatile("" ::: "memory");   // keep LDS reads below the wait
#else
  asm volatile("s_wait_asynccnt 0x0" ::: "memory");
#endif
}

// Straight-line unroll with constant offsets: one VGPR address pair per
// segment, per-iteration delta (128 B) carried in the instruction immediate.
template <int K, int FULL>
__device__ __forceinline__ void stage_rec(const float* g, float* l) {
  if constexpr (K < FULL) {
    async_g2l_b32_off<K * 128>(g, l);
    stage_rec<K + 1, FULL>(g, l);
  }
}

// Stage N contiguous floats (one segment of a sample) into LDS.
template <int N>
__device__ __forceinline__ void stage_seg(const float* __restrict__ g,
                                          float* l, int lane) {
  constexpr int FULL = N / 32;   // full 32-lane waves of 128 B
  constexpr int REM  = N % 32;   // lane-predicated tail
  stage_rec<0, FULL>(g + lane, l + lane);
  if (REM > 0 && lane < REM)
    async_g2l_b32_off<FULL * 128>(g + lane, l + lane);
}

// ---------------- kernel ----------------
// One wave32 per sample. lane == embedding column (0..31).
// Stage the sample's 1980 floats (user||ctx||doc, each segment contiguous)
// into this wave's LDS buffer with async copies, then reduce out of LDS.
__global__ __launch_bounds__(BLOCK_THREADS) void FM_model_30408368456213_kernel(
    const float* __restrict__ user, const float* __restrict__ ctx,
    const float* __restrict__ doc, float* __restrict__ out, int n) {
  __shared__ float buf[WAVES_PER_BLOCK][SAMPLE_FLOATS];  // 31.7 KB / block

  const int lane = threadIdx.x & 31;
  const int w    = threadIdx.x >> 5;
  const int s    = blockIdx.x * WAVES_PER_BLOCK + w;
  if (s >= n) return;

  float* lbuf = &buf[w][0];

  // ---- async stage: HBM -> LDS, 63 straight-line 128 B copies ----
  stage_seg<NF_USER * ROW>(user + (size_t)s * (NF_USER * ROW), lbuf, lane);
  stage_seg<NF_CTX * ROW>(ctx + (size_t)s * (NF_CTX * ROW),
                          lbuf + NF_USER * ROW, lane);
  stage_seg<NF_DOC * ROW>(doc + (size_t)s * (NF_DOC * ROW),
                          lbuf + (NF_USER + NF_CTX) * ROW, lane);
  wait_async0();

  // ---- FM reduction: lane c accumulates S_c and SQ_c over 60 fields ----
  float sv = 0.0f, sv2 = 0.0f;
#pragma unroll
  for (int f = 0; f < NF; ++f) {
    float v = lbuf[f * ROW + lane];
    sv += v;
    sv2 = fmaf(v, v, sv2);
  }

  // linear terms: element 32 of each of the 60 rows, spread across lanes
  float lin = lbuf[lane * ROW + VL];
  if (lane < NF - 32) lin += lbuf[(lane + 32) * ROW + VL];

  // per-lane partial of  0.5*sum_c(S_c^2 - SQ_c) + sum_f lin_f
  float p = 0.5f * (sv * sv - sv2) + lin;

  // wave32 butterfly reduction
#pragma unroll
  for (int m = 16; m >= 1; m >>= 1) p += __shfl_xor(p, m, 32);

  if (lane == 0) out[s] = 1.0f / (1.0f + expf(-p));
}

// ---------------- launcher ----------------
extern "C" void kernel_launch(void* const* d_in, const int* in_sizes, int n_in,
                              void* d_out, int out_size, void* d_ws, size_t ws_size,
                              hipStream_t stream) {
  (void)in_sizes; (void)n_in; (void)d_ws; (void)ws_size;
  const float* user = (const float*)d_in[0];
  const float* ctx  = (const float*)d_in[1];
  const float* doc  = (const float*)d_in[2];
  float* out = (float*)d_out;

  const int n = out_size;  // 16384 samples
  const int blocks = (n + WAVES_PER_BLOCK - 1) / WAVES_PER_BLOCK;
  hipLaunchKernelGGL(FM_model_30408368456213_kernel, dim3(blocks),
                     dim3(BLOCK_THREADS), 0, stream, user, ctx, doc, out, n);
}